// Attention_9251359556109
// MI455X (gfx1250) — compile-verified
//
#include <hip/hip_runtime.h>
#include <stdint.h>

// Problem constants (match reference)
#define NB   64          // batch
#define NHQ  32          // query heads
#define NHKV 8           // kv heads
#define DH   128         // head dim
#define BS   16          // keys per paged block
#define MAXB 128         // max blocks per sequence
#define GRP  4           // query heads per kv head
#define NW   4           // waves per workgroup
#define QK_SCALE 0.08838834764831845f

typedef __attribute__((ext_vector_type(2))) float v2f;
typedef __attribute__((ext_vector_type(8))) float v8f;
typedef __attribute__((ext_vector_type(4))) uint32_t v4u;
typedef __attribute__((ext_vector_type(8))) uint32_t v8u;

// ---- LDS byte offset of a shared-memory pointer ----
__device__ __forceinline__ uint32_t lds_byte_off(const void* p) {
  return (uint32_t)(uintptr_t)p;   // low 32 bits of LDS-aperture address = WG offset
}

// ---- CDNA5 Tensor Data Mover: one-instruction 2D tile copy (TENSORcnt path) ----
// Copies a 16x128 f32 tile whose rows are 1024 floats apart (paged KV layout).
__device__ __forceinline__ void tdm_load_tile_16x128(uint32_t lds_off, const float* gsrc) {
  const uint64_t ga = (uint64_t)(uintptr_t)gsrc;
  v4u g0; v8u g1;
  v4u g2 = (v4u)(0u);
  v4u g3 = (v4u)(0u);
  g0[0] = 1u;                                        // count=1, is_restore=0, gather off
  g0[1] = lds_off;                                   // lds_addr (bytes)
  g0[2] = (uint32_t)ga;                              // global_addr[31:0]
  g0[3] = (uint32_t)((ga >> 32) & 0x01FFFFFFu)       // global_addr[56:32]
        | (2u << 30);                                // type = 2 ("image")
  g1[0] = 0x00020000u;                               // workgroup_mask=0, data_size=2 (4B)
  g1[1] = 128u << 16;                                // tensor_dim0 = 128 elements
  g1[2] = 16u << 16;                                 // tensor_dim1 = 16 rows
  g1[3] = 128u << 16;                                // tile_dim0 = 128
  g1[4] = 16u;                                       // tile_dim1 = 16, tile_dim2 = 0
  g1[5] = 1024u;                                     // tensor_dim0_stride = 1024 elements
  g1[6] = 0u;
  g1[7] = 0u;
  asm volatile("tensor_load_to_lds %0, %1, %2, %3"
               :: "s"(g0), "s"(g1), "s"(g2), "s"(g3) : "memory");
}

// ---- CDNA5 async global->LDS copy (ASYNCcnt path) ----
__device__ __forceinline__ void async_copy_b128(uint32_t lds_off, const float* gsrc) {
  asm volatile("global_load_async_to_lds_b128 %0, %1, off"
               :: "v"(lds_off), "v"(gsrc) : "memory");
}
__device__ __forceinline__ void wait_async0() {
  asm volatile("s_wait_asynccnt 0x0" ::: "memory");
}
__device__ __forceinline__ void wait_ds0() {
  asm volatile("s_wait_dscnt 0x0" ::: "memory");
}

__global__ __launch_bounds__(NW * 32, 1)
void paged_gqa_decode_kernel(const float* __restrict__ q,
                             const float* __restrict__ kc,
                             const float* __restrict__ vc,
                             const int*   __restrict__ bt,
                             const int*   __restrict__ cl,
                             float*       __restrict__ out) {
  __shared__ float tile[NW][BS * DH];      // 32 KB: K tile, then reused for V tile
  __shared__ float p_lds[NW][16 * 16];     // 4 KB: P in WMMA-A layout (rows 4..15 = 0)
  __shared__ float o_mrg[NW][GRP][DH];     // 8 KB: per-wave partial output
  __shared__ float m_mrg[NW][GRP];
  __shared__ float l_mrg[NW][GRP];

  const int h    = blockIdx.x;             // kv head
  const int b    = blockIdx.y;             // batch
  const int tid  = threadIdx.x;
  const int w    = tid >> 5;               // wave id (wave32)
  const int lane = tid & 31;
  const int key  = lane & 15;              // N index / M index inside 16-lane half
  const int half = lane >> 4;              // K-half select for WMMA layouts

  const int ctx  = cl[b];
  const int nblk = (ctx + BS - 1) / BS;

  // zero P scratch (rows 4..15 must stay zero forever)
  #pragma unroll
  for (int i = 0; i < 8; ++i) p_lds[w][lane + 32 * i] = 0.0f;

  // ---- preload Q in WMMA 32-bit A layout: lane m, VGPR pair = dims (4k+2h, 4k+2h+1)
  float qa[64];
  {
    const float* qrow = q + ((size_t)b * NHQ + (size_t)h * GRP + key) * DH;
    #pragma unroll
    for (int kk = 0; kk < 32; ++kk) {
      if (key < GRP) {
        float2 t2 = *(const float2*)(qrow + 4 * kk + 2 * half);
        qa[2 * kk]     = t2.x * QK_SCALE;
        qa[2 * kk + 1] = t2.y * QK_SCALE;
      } else {
        qa[2 * kk] = 0.0f;
        qa[2 * kk + 1] = 0.0f;
      }
    }
  }

  // flash-attention state
  v8f acc[8];
  #pragma unroll
  for (int t = 0; t < 8; ++t)
    #pragma unroll
    for (int j = 0; j < 8; ++j) acc[t][j] = 0.0f;
  float m_run[GRP], l_run[GRP];
  #pragma unroll
  for (int g = 0; g < GRP; ++g) { m_run[g] = -__builtin_inff(); l_run[g] = 0.0f; }

  const uint32_t tile_off = lds_byte_off(tile[w]);

  for (int i = w; i < nblk; i += NW) {
    const int pb = bt[b * MAXB + i];                       // wave-uniform page
    const float* kbase = kc + ((size_t)pb * BS * NHKV + h) * DH;
    const float* vbase = vc + ((size_t)pb * BS * NHKV + h) * DH;

    // ---- stage K tile with ONE Tensor Data Mover descriptor (16x128 f32, stride 1024)
    tdm_load_tile_16x128(tile_off, kbase);
    // warm V lines while the TDM copy is in flight
    __builtin_prefetch(vbase + (size_t)(lane >> 1) * (NHKV * DH) + (lane & 1) * 64, 0, 1);
    __builtin_amdgcn_s_wait_tensorcnt(0);                  // K tile resident in LDS

    // ---- scores C[16x16] = Q * K^T via 32 x v_wmma_f32_16x16x4_f32
    v8f c;
    #pragma unroll
    for (int j = 0; j < 8; ++j) c[j] = 0.0f;
    const float* krow = tile[w] + key * DH + 2 * half;
    #pragma unroll
    for (int kk = 0; kk < 32; ++kk) {
      float2 kb = *(const float2*)(krow + 4 * kk);
      v2f a; a[0] = qa[2 * kk]; a[1] = qa[2 * kk + 1];
      v2f bb; bb[0] = kb.x; bb[1] = kb.y;
      c = __builtin_amdgcn_wmma_f32_16x16x4_f32(false, a, false, bb,
                                                (short)0, c, false, false);
    }

    wait_ds0();                       // K-tile LDS reads done before overwrite
    // ---- stage V tile into same buffer via async engine (overlaps softmax below)
    {
      const float* gsrc = vbase + lane * 4;
      #pragma unroll
      for (int r = 0; r < BS; ++r)
        async_copy_b128(tile_off + (uint32_t)(r * DH * 4 + lane * 16),
                        gsrc + (size_t)r * (NHKV * DH));
    }

    // ---- online softmax on lanes 0..15 (key = lane)
    const int kidx = i * BS + key;
    const bool valid = kidx < ctx;
    float alpha[GRP], pval[GRP];
    #pragma unroll
    for (int g = 0; g < GRP; ++g) {
      float s = valid ? c[g] : -__builtin_inff();
      float mb = s;
      mb = fmaxf(mb, __shfl_xor(mb, 8, 16));
      mb = fmaxf(mb, __shfl_xor(mb, 4, 16));
      mb = fmaxf(mb, __shfl_xor(mb, 2, 16));
      mb = fmaxf(mb, __shfl_xor(mb, 1, 16));
      float m_new = fmaxf(m_run[g], mb);
      float p = valid ? __expf(s - m_new) : 0.0f;
      float ps = p;
      ps += __shfl_xor(ps, 8, 16);
      ps += __shfl_xor(ps, 4, 16);
      ps += __shfl_xor(ps, 2, 16);
      ps += __shfl_xor(ps, 1, 16);
      alpha[g]  = (m_run[g] == -__builtin_inff()) ? 0.0f : __expf(m_run[g] - m_new);
      l_run[g]  = l_run[g] * alpha[g] + ps;
      m_run[g]  = m_new;
      pval[g]   = p;
    }
    if (lane < 16) {
      #pragma unroll
      for (int g = 0; g < GRP; ++g) p_lds[w][g * 16 + lane] = pval[g];
    }
    // rescale accumulators (rows 0..3 on lanes 0..15 are the live data)
    #pragma unroll
    for (int t = 0; t < 8; ++t)
      #pragma unroll
      for (int g = 0; g < GRP; ++g) acc[t][g] *= alpha[g];

    wait_async0();                    // V tile ready

    // ---- A operands for P*V (depend only on key-chunk, reuse across d-tiles)
    v2f pa[4];
    #pragma unroll
    for (int kk2 = 0; kk2 < 4; ++kk2) {
      float2 t2 = *(const float2*)&p_lds[w][key * 16 + 4 * kk2 + 2 * half];
      pa[kk2][0] = t2.x; pa[kk2][1] = t2.y;
    }
    // ---- acc += P * V : 8 d-tiles x 4 key-chunks of WMMA
    #pragma unroll
    for (int t = 0; t < 8; ++t) {
      #pragma unroll
      for (int kk2 = 0; kk2 < 4; ++kk2) {
        const int kr = 4 * kk2 + 2 * half;
        v2f bb;
        bb[0] = tile[w][(kr    ) * DH + t * 16 + key];
        bb[1] = tile[w][(kr + 1) * DH + t * 16 + key];
        acc[t] = __builtin_amdgcn_wmma_f32_16x16x4_f32(false, pa[kk2], false, bb,
                                                       (short)0, acc[t], false, false);
      }
    }
    wait_ds0();                       // V-tile reads done before next K staging
  }

  // ---- per-wave partials -> LDS
  if (lane < 16) {
    #pragma unroll
    for (int g = 0; g < GRP; ++g)
      #pragma unroll
      for (int t = 0; t < 8; ++t)
        o_mrg[w][g][t * 16 + lane] = acc[t][g];
  }
  if (lane == 0) {
    #pragma unroll
    for (int g = 0; g < GRP; ++g) { m_mrg[w][g] = m_run[g]; l_mrg[w][g] = l_run[g]; }
  }
  __syncthreads();

  // ---- merge 4 waves, write output (512 elems / 128 threads = 4 each)
  for (int e = tid; e < GRP * DH; e += NW * 32) {
    const int g = e >> 7;
    const int d = e & (DH - 1);
    float mstar = -__builtin_inff();
    #pragma unroll
    for (int ww = 0; ww < NW; ++ww) mstar = fmaxf(mstar, m_mrg[ww][g]);
    float num = 0.0f, den = 0.0f;
    #pragma unroll
    for (int ww = 0; ww < NW; ++ww) {
      const float mw = m_mrg[ww][g];
      const float cw = (mw == -__builtin_inff()) ? 0.0f : __expf(mw - mstar);
      num += cw * o_mrg[ww][g][d];
      den += cw * l_mrg[ww][g];
    }
    out[(((size_t)b * NHQ) + (size_t)h * GRP + g) * DH + d] = num / den;
  }
}

extern "C" void kernel_launch(void* const* d_in, const int* in_sizes, int n_in,
                              void* d_out, int out_size, void* d_ws, size_t ws_size,
                              hipStream_t stream) {
  const float* q  = (const float*)d_in[0];
  const float* kc = (const float*)d_in[1];
  const float* vc = (const float*)d_in[2];
  const int*   bt = (const int*)d_in[3];
  const int*   cl = (const int*)d_in[4];
  float* out = (float*)d_out;

  dim3 grid(NHKV, NB);     // 8 x 64 = 512 workgroups: one per (kv-head, batch)
  dim3 block(NW * 32);     // 4 waves (wave32)
  hipLaunchKernelGGL(paged_gqa_decode_kernel, grid, block, 0, stream,
                     q, kc, vc, bt, cl, out);
}